// HypergraphRayleighQuotientLoss_20710332302128
// MI455X (gfx1250) — compile-verified
//
#include <hip/hip_runtime.h>
#include <hip/hip_bf16.h>

typedef float v2f __attribute__((ext_vector_type(2)));
typedef float v8f __attribute__((ext_vector_type(8)));

#define K_COLS 16

// ---------------------------------------------------------------------------
// Pass 1: degrees.  Dv[n] += w[e], De[e] += 1 for every pin.
// ---------------------------------------------------------------------------
__global__ void degree_kernel(const int* __restrict__ node_idx,
                              const int* __restrict__ edge_idx,
                              const float* __restrict__ w,
                              float* __restrict__ Dv,
                              float* __restrict__ De,
                              long long nnz) {
    long long i = (long long)blockIdx.x * blockDim.x + threadIdx.x;
    long long stride = (long long)gridDim.x * blockDim.x;
    for (; i < nnz; i += stride) {
        int n = node_idx[i];
        int e = edge_idx[i];
        atomicAdd(&Dv[n], w[e]);
        atomicAdd(&De[e], 1.0f);
    }
}

// ---------------------------------------------------------------------------
// Pass 2 (hot): per pin, scatter raw Z row into z_sum[e][:] and accumulate
// acc1[k] = sum_pins w[e] * Z[n][k]^2  (raw cut quadratic term).
// Z rows (64B) and z_sum (6.4MB) are L2-resident; bound by f32 atomics.
// ---------------------------------------------------------------------------
__global__ void pin_kernel(const int* __restrict__ node_idx,
                           const int* __restrict__ edge_idx,
                           const float* __restrict__ w,
                           const float* __restrict__ Z,
                           float* __restrict__ zsum,
                           float* __restrict__ acc1,
                           long long nnz) {
    float s[K_COLS];
#pragma unroll
    for (int k = 0; k < K_COLS; ++k) s[k] = 0.0f;

    long long i = (long long)blockIdx.x * blockDim.x + threadIdx.x;
    long long stride = (long long)gridDim.x * blockDim.x;
    for (; i < nnz; i += stride) {
        int n = node_idx[i];
        int e = edge_idx[i];
        float we = w[e];
        const float4* zr = (const float4*)(Z + (size_t)n * K_COLS);
        float4 z0 = zr[0], z1 = zr[1], z2 = zr[2], z3 = zr[3];
        float zv[K_COLS] = {z0.x, z0.y, z0.z, z0.w,
                            z1.x, z1.y, z1.z, z1.w,
                            z2.x, z2.y, z2.z, z2.w,
                            z3.x, z3.y, z3.z, z3.w};
        float* zs = zsum + (size_t)e * K_COLS;
#pragma unroll
        for (int k = 0; k < K_COLS; ++k) {
            atomicAdd(&zs[k], zv[k]);
            s[k] = fmaf(we, zv[k] * zv[k], s[k]);
        }
    }

    // wave32 butterfly reduction of the 16-vector
#pragma unroll
    for (int off = 16; off > 0; off >>= 1) {
#pragma unroll
        for (int k = 0; k < K_COLS; ++k)
            s[k] += __shfl_xor(s[k], off);
    }

    __shared__ float sm[K_COLS];
    if (threadIdx.x < K_COLS) sm[threadIdx.x] = 0.0f;
    __syncthreads();
    if ((threadIdx.x & 31) == 0) {
#pragma unroll
        for (int k = 0; k < K_COLS; ++k) atomicAdd(&sm[k], s[k]);
    }
    __syncthreads();
    if (threadIdx.x < K_COLS) atomicAdd(&acc1[threadIdx.x], sm[threadIdx.x]);
}

// ---------------------------------------------------------------------------
// Pass 3: node column stats via WMMA.  C += ones(16x4) @ B(4x16) makes every
// row of C the column sums of the tile.  Two accumulators:
//   colsq[k]  = sum_i Z[i][k]^2          (for the column norms)
//   rawexp[k] = sum_i clamp(Dv[i]) * Z[i][k]^2   (expansion numerator)
// B layout (32-bit, 4x16): lanes0-15 hold N=lane, {VGPR0,VGPR1}={K0,K1};
// lanes16-31 hold N=lane-16, {K2,K3}.  Sum over K is order-invariant.
// ---------------------------------------------------------------------------
__global__ void node_stats_kernel(const float* __restrict__ Z,
                                  const float* __restrict__ Dv,
                                  float* __restrict__ colsq,
                                  float* __restrict__ rawexp,
                                  int n_nodes) {
    const int lane = threadIdx.x & 31;
    const int gwave = (blockIdx.x * blockDim.x + threadIdx.x) >> 5;
    const int nwaves = (gridDim.x * blockDim.x) >> 5;
    const int nchunks = (n_nodes + 3) >> 2;
    const int n = lane & 15;
    const int rsel = (lane >> 4) << 1;  // 0 or 2

    v8f c_sq = {};
    v8f c_dv = {};
    v2f a;
    a.x = 1.0f; a.y = 1.0f;

    for (int ch = gwave; ch < nchunks; ch += nwaves) {  // wave-uniform loop
        int r0 = ch * 4 + rsel;
        int r1 = r0 + 1;
        float z0 = 0.0f, z1 = 0.0f, d0 = 0.0f, d1 = 0.0f;
        if (r0 < n_nodes) {
            z0 = Z[(size_t)r0 * K_COLS + n];
            d0 = Dv[r0]; d0 = (d0 == 0.0f) ? 1.0f : d0;
        }
        if (r1 < n_nodes) {
            z1 = Z[(size_t)r1 * K_COLS + n];
            d1 = Dv[r1]; d1 = (d1 == 0.0f) ? 1.0f : d1;
        }
        v2f b_sq; b_sq.x = z0 * z0;      b_sq.y = z1 * z1;
        v2f b_dv; b_dv.x = d0 * z0 * z0; b_dv.y = d1 * z1 * z1;
        c_sq = __builtin_amdgcn_wmma_f32_16x16x4_f32(
            false, a, false, b_sq, (short)0, c_sq, false, false);
        c_dv = __builtin_amdgcn_wmma_f32_16x16x4_f32(
            false, a, false, b_dv, (short)0, c_dv, false, false);
    }
    // Row M=0 of C lives in VGPR0 of lanes 0..15 (N = lane).
    if (lane < 16) {
        atomicAdd(&colsq[n], c_sq[0]);
        atomicAdd(&rawexp[n], c_dv[0]);
    }
}

// ---------------------------------------------------------------------------
// Pass 4: edge cut quadratic term via WMMA column reduction:
//   acc2[k] = sum_e w[e] * z_sum_raw[e][k]^2 / clamp(De[e])
// ---------------------------------------------------------------------------
__global__ void edge_cut_kernel(const float* __restrict__ zsum,
                                const float* __restrict__ De,
                                const float* __restrict__ w,
                                float* __restrict__ acc2,
                                int n_edges) {
    const int lane = threadIdx.x & 31;
    const int gwave = (blockIdx.x * blockDim.x + threadIdx.x) >> 5;
    const int nwaves = (gridDim.x * blockDim.x) >> 5;
    const int nchunks = (n_edges + 3) >> 2;
    const int n = lane & 15;
    const int rsel = (lane >> 4) << 1;

    v8f c = {};
    v2f a;
    a.x = 1.0f; a.y = 1.0f;

    for (int ch = gwave; ch < nchunks; ch += nwaves) {  // wave-uniform loop
        int e0 = ch * 4 + rsel;
        int e1 = e0 + 1;
        float v0 = 0.0f, v1 = 0.0f;
        if (e0 < n_edges) {
            float d = De[e0]; d = (d == 0.0f) ? 1.0f : d;
            float s = zsum[(size_t)e0 * K_COLS + n];
            v0 = w[e0] * s * s / d;
        }
        if (e1 < n_edges) {
            float d = De[e1]; d = (d == 0.0f) ? 1.0f : d;
            float s = zsum[(size_t)e1 * K_COLS + n];
            v1 = w[e1] * s * s / d;
        }
        v2f b; b.x = v0; b.y = v1;
        c = __builtin_amdgcn_wmma_f32_16x16x4_f32(
            false, a, false, b, (short)0, c, false, false);
    }
    if (lane < 16) atomicAdd(&acc2[n], c[0]);
}

// ---------------------------------------------------------------------------
// Pass 5: finalize.  norm2[k] = max(colsq,1e-24) (== max(sqrt,1e-12)^2).
//   expansion[k] = rawexp/norm2
//   cut[k]       = 0.5*(acc1 - acc2)/norm2
//   loss = mean(expansion) - mean(cut)
// ---------------------------------------------------------------------------
__global__ void finalize_kernel(const float* __restrict__ colsq,
                                const float* __restrict__ rawexp,
                                const float* __restrict__ acc1,
                                const float* __restrict__ acc2,
                                float* __restrict__ out) {
    int k = threadIdx.x;
    float v = 0.0f;
    if (k < K_COLS) {
        float n2 = fmaxf(colsq[k], 1e-24f);
        float expn = rawexp[k] / n2;
        float cut = 0.5f * (acc1[k] - acc2[k]) / n2;
        v = (expn - cut) * (1.0f / (float)K_COLS);
    }
#pragma unroll
    for (int off = 16; off > 0; off >>= 1) v += __shfl_xor(v, off);
    if (k == 0) out[0] = v;
}

// ---------------------------------------------------------------------------
extern "C" void kernel_launch(void* const* d_in, const int* in_sizes, int n_in,
                              void* d_out, int out_size, void* d_ws, size_t ws_size,
                              hipStream_t stream) {
    const float* Z   = (const float*)d_in[0];
    const int*   hid = (const int*)d_in[1];
    const float* w   = (const float*)d_in[3];

    const int n_nodes = in_sizes[0] / K_COLS;
    const long long nnz = in_sizes[1] / 2;
    const int n_edges = in_sizes[3];

    const int* node_idx = hid;
    const int* edge_idx = hid + nnz;

    // Workspace layout (contiguous floats)
    float* Dv     = (float*)d_ws;
    float* De     = Dv + n_nodes;
    float* zsum   = De + n_edges;
    float* colsq  = zsum + (size_t)n_edges * K_COLS;
    float* rawexp = colsq + K_COLS;
    float* acc1   = rawexp + K_COLS;
    float* acc2   = acc1 + K_COLS;

    size_t zero_bytes =
        sizeof(float) * ((size_t)n_nodes + (size_t)n_edges +
                         (size_t)n_edges * K_COLS + 4 * K_COLS);
    hipMemsetAsync(d_ws, 0, zero_bytes, stream);

    const int block = 256;
    long long pin_blocks_ll = (nnz + block - 1) / block;
    int pin_blocks = (pin_blocks_ll > 25000) ? 25000 : (int)pin_blocks_ll;

    degree_kernel<<<pin_blocks, block, 0, stream>>>(node_idx, edge_idx, w,
                                                    Dv, De, nnz);
    pin_kernel<<<pin_blocks, block, 0, stream>>>(node_idx, edge_idx, w, Z,
                                                 zsum, acc1, nnz);

    int node_chunks = (n_nodes + 3) / 4;
    int node_waves_blocks = (node_chunks + 7) / 8;  // 8 waves per 256-thd block
    if (node_waves_blocks > 2048) node_waves_blocks = 2048;
    node_stats_kernel<<<node_waves_blocks, block, 0, stream>>>(
        Z, Dv, colsq, rawexp, n_nodes);

    int edge_chunks = (n_edges + 3) / 4;
    int edge_waves_blocks = (edge_chunks + 7) / 8;
    if (edge_waves_blocks > 2048) edge_waves_blocks = 2048;
    edge_cut_kernel<<<edge_waves_blocks, block, 0, stream>>>(
        zsum, De, w, acc2, n_edges);

    finalize_kernel<<<1, 32, 0, stream>>>(colsq, rawexp, acc1, acc2,
                                          (float*)d_out);
}